// SharedTransitionUp_56710748176531
// MI455X (gfx1250) — compile-verified
//
#include <hip/hip_runtime.h>

#define B_    4
#define N1_   16384
#define N2_   4096
#define C1_   128
#define C2_   256
#define CIN_  384
#define OUT_  128
#define ROWS_ (B_ * N1_)      /* 65536 */
#define EPS_BN 1e-5f

typedef __attribute__((ext_vector_type(2))) float v2f;
typedef __attribute__((ext_vector_type(8))) float v8f;

// ---- workspace layout (in floats) ----
#define I_INTERP ((size_t)0)
#define I_H      ((size_t)ROWS_ * C2_)                 // 16.7M floats
#define I_SUM    (I_H + (size_t)ROWS_ * OUT_)          // +8.4M floats
#define I_SUMSQ  (I_SUM + 128)
#define I_SCALE  (I_SUMSQ + 128)
#define I_SHIFT  (I_SCALE + 128)

// ---------------------------------------------------------------------------
// Kernel 0: zero the BN accumulators (sum[128] + sumsq[128])
// ---------------------------------------------------------------------------
__global__ void zero_sums_kernel(float* __restrict__ p) {
    p[threadIdx.x] = 0.0f;
}

// ---------------------------------------------------------------------------
// Kernel 1: 3-NN over xyz2 (staged in LDS) + inverse-distance interpolation
// grid = B * (N1/256) blocks, 256 threads. Thread = one query point for KNN,
// then thread = one channel for the cooperative coalesced gather.
// ---------------------------------------------------------------------------
__global__ __launch_bounds__(256) void knn_interp_kernel(
    const float* __restrict__ xyz1, const float* __restrict__ xyz2,
    const float* __restrict__ points2, float* __restrict__ interp)
{
    __shared__ float s_xyz2[N2_ * 3];   // 48 KB
    __shared__ float s_w[256 * 3];
    __shared__ int   s_i[256 * 3];

    const int tid    = threadIdx.x;
    const int b      = blockIdx.x >> 6;          // 64 blocks per batch
    const int n1base = (blockIdx.x & 63) << 8;

    // stage this batch's xyz2 into LDS (coalesced)
    const float* x2 = xyz2 + (size_t)b * N2_ * 3;
    for (int i = tid; i < N2_ * 3; i += 256) s_xyz2[i] = x2[i];
    __syncthreads();

    // per-thread top-3 scan (LDS reads are wave-uniform -> broadcast)
    const int n1 = n1base + tid;
    const float* q = xyz1 + ((size_t)b * N1_ + n1) * 3;
    const float qx = q[0], qy = q[1], qz = q[2];

    float d0 = 3.4e38f, d1 = 3.4e38f, d2 = 3.4e38f;
    int   i0 = 0, i1 = 0, i2 = 0;
    for (int j = 0; j < N2_; ++j) {
        float dx = qx - s_xyz2[j * 3 + 0];
        float dy = qy - s_xyz2[j * 3 + 1];
        float dz = qz - s_xyz2[j * 3 + 2];
        float dd = dx * dx + dy * dy + dz * dz;
        if (dd < d0)      { d2 = d1; i2 = i1; d1 = d0; i1 = i0; d0 = dd; i0 = j; }
        else if (dd < d1) { d2 = d1; i2 = i1; d1 = dd; i1 = j; }
        else if (dd < d2) { d2 = dd; i2 = j; }
    }
    float w0 = 1.0f / (sqrtf(fmaxf(d0, 0.0f)) + 1e-8f);
    float w1 = 1.0f / (sqrtf(fmaxf(d1, 0.0f)) + 1e-8f);
    float w2 = 1.0f / (sqrtf(fmaxf(d2, 0.0f)) + 1e-8f);
    float inv = 1.0f / (w0 + w1 + w2);
    s_w[tid * 3 + 0] = w0 * inv;
    s_w[tid * 3 + 1] = w1 * inv;
    s_w[tid * 3 + 2] = w2 * inv;
    s_i[tid * 3 + 0] = i0;
    s_i[tid * 3 + 1] = i1;
    s_i[tid * 3 + 2] = i2;
    __syncthreads();

    // cooperative interpolation: thread = channel c, loop over block's points.
    // All points2 row reads are fully coalesced (256 lanes x 4B contiguous).
    const float* p2 = points2 + (size_t)b * N2_ * C2_;
    const int c = tid;
    for (int p = 0; p < 256; ++p) {
        const float a0 = s_w[p * 3 + 0], a1 = s_w[p * 3 + 1], a2 = s_w[p * 3 + 2];
        const int   j0 = s_i[p * 3 + 0], j1 = s_i[p * 3 + 1], j2 = s_i[p * 3 + 2];
        float v = a0 * p2[(size_t)j0 * C2_ + c]
                + a1 * p2[(size_t)j1 * C2_ + c]
                + a2 * p2[(size_t)j2 * C2_ + c];
        interp[((size_t)(b * N1_ + n1base + p)) * C2_ + c] = v;
    }
}

// ---------------------------------------------------------------------------
// Kernel 2: GEMM1 (65536x384 @ 384x128) via V_WMMA_F32_16X16X4_F32
// + bias + per-channel sum/sumsq reduction for BatchNorm.
// 256 threads = 8 waves; each wave computes a 16(M) x 128(N) strip.
// A row k<128 comes from points1, k>=128 from interp.
// ---------------------------------------------------------------------------
__global__ __launch_bounds__(256) void gemm1_bn_kernel(
    const float* __restrict__ points1, const float* __restrict__ interp,
    const float* __restrict__ w1, const float* __restrict__ b1,
    float* __restrict__ H, float* __restrict__ gsum, float* __restrict__ gsumsq)
{
    __shared__ float s_sum[128];
    __shared__ float s_sq[128];

    const int tid = threadIdx.x;
    if (tid < 128) { s_sum[tid] = 0.0f; s_sq[tid] = 0.0f; }
    __syncthreads();

    const int wave    = tid >> 5;
    const int lane    = tid & 31;
    const int rowbase = (blockIdx.x * 8 + wave) * 16;
    const int m       = lane & 15;           // A: M index / B: N index
    const int kh      = (lane >> 4) * 2;     // lanes 16-31 hold K=2,3
    const int r       = rowbase + m;

    v8f acc[8];
    v8f zero = {};
    for (int nt = 0; nt < 8; ++nt) acc[nt] = zero;

    for (int k0 = 0; k0 < CIN_; k0 += 4) {
        const int k = k0 + kh;
        v2f a;
        if (k0 < C1_) a = *(const v2f*)(points1 + (size_t)r * C1_ + k);
        else          a = *(const v2f*)(interp  + (size_t)r * C2_ + (k - C1_));
        for (int nt = 0; nt < 8; ++nt) {
            const int n = nt * 16 + m;
            v2f bf = *(const v2f*)(w1 + (size_t)n * CIN_ + k);
            acc[nt] = __builtin_amdgcn_wmma_f32_16x16x4_f32(
                false, a, false, bf, (short)0, acc[nt], false, false);
        }
    }

    // epilogue: bias, store H, reduce sum/sumsq per channel
    const int rowoff = (lane >> 4) * 8;      // C/D: lanes 16-31 hold M=8..15
    for (int nt = 0; nt < 8; ++nt) {
        const int o = nt * 16 + m;
        const float bias = b1[o];
        float psum = 0.0f, psq = 0.0f;
        for (int v = 0; v < 8; ++v) {
            float val = acc[nt][v] + bias;
            H[(size_t)(rowbase + rowoff + v) * OUT_ + o] = val;
            psum += val;
            psq  += val * val;
        }
        atomicAdd(&s_sum[o], psum);
        atomicAdd(&s_sq[o],  psq);
    }
    __syncthreads();
    if (tid < 128) {
        atomicAdd(&gsum[tid],   s_sum[tid]);
        atomicAdd(&gsumsq[tid], s_sq[tid]);
    }
}

// ---------------------------------------------------------------------------
// Kernel 3: fold global sums into BN scale/shift (training-mode batch stats)
// ---------------------------------------------------------------------------
__global__ void bn_stats_kernel(
    const float* __restrict__ gsum, const float* __restrict__ gsumsq,
    const float* __restrict__ gamma, const float* __restrict__ beta,
    float* __restrict__ scale, float* __restrict__ shift)
{
    const int o = threadIdx.x;
    const float invn = 1.0f / (float)ROWS_;
    float mean = gsum[o] * invn;
    float var  = fmaxf(gsumsq[o] * invn - mean * mean, 0.0f);
    float sc   = gamma[o] * rsqrtf(var + EPS_BN);
    scale[o] = sc;
    shift[o] = beta[o] - mean * sc;
}

// ---------------------------------------------------------------------------
// Kernel 4: normalize + ReLU fused into A-fragment load, then
// GEMM2 (65536x128 @ 128x128) via V_WMMA_F32_16X16X4_F32 -> d_out
// ---------------------------------------------------------------------------
__global__ __launch_bounds__(256) void gemm2_kernel(
    const float* __restrict__ H, const float* __restrict__ scale,
    const float* __restrict__ shift, const float* __restrict__ w2,
    const float* __restrict__ b2, float* __restrict__ out)
{
    __shared__ float s_scale[128];
    __shared__ float s_shift[128];
    const int tid = threadIdx.x;
    if (tid < 128) { s_scale[tid] = scale[tid]; s_shift[tid] = shift[tid]; }
    __syncthreads();

    const int wave    = tid >> 5;
    const int lane    = tid & 31;
    const int rowbase = (blockIdx.x * 8 + wave) * 16;
    const int m       = lane & 15;
    const int kh      = (lane >> 4) * 2;
    const int r       = rowbase + m;

    v8f acc[8];
    v8f zero = {};
    for (int nt = 0; nt < 8; ++nt) acc[nt] = zero;

    for (int k0 = 0; k0 < OUT_; k0 += 4) {
        const int k = k0 + kh;
        v2f hv = *(const v2f*)(H + (size_t)r * OUT_ + k);
        v2f a;
        a.x = fmaxf(hv.x * s_scale[k]     + s_shift[k],     0.0f);
        a.y = fmaxf(hv.y * s_scale[k + 1] + s_shift[k + 1], 0.0f);
        for (int nt = 0; nt < 8; ++nt) {
            const int n = nt * 16 + m;
            v2f bf = *(const v2f*)(w2 + (size_t)n * OUT_ + k);
            acc[nt] = __builtin_amdgcn_wmma_f32_16x16x4_f32(
                false, a, false, bf, (short)0, acc[nt], false, false);
        }
    }

    const int rowoff = (lane >> 4) * 8;
    for (int nt = 0; nt < 8; ++nt) {
        const int o = nt * 16 + m;
        const float bias = b2[o];
        for (int v = 0; v < 8; ++v) {
            out[(size_t)(rowbase + rowoff + v) * OUT_ + o] = acc[nt][v] + bias;
        }
    }
}

// ---------------------------------------------------------------------------
extern "C" void kernel_launch(void* const* d_in, const int* in_sizes, int n_in,
                              void* d_out, int out_size, void* d_ws, size_t ws_size,
                              hipStream_t stream)
{
    (void)in_sizes; (void)n_in; (void)out_size; (void)ws_size;

    const float* xyz1    = (const float*)d_in[0];
    const float* xyz2    = (const float*)d_in[1];
    const float* points1 = (const float*)d_in[2];
    const float* points2 = (const float*)d_in[3];
    const float* w1      = (const float*)d_in[4];
    const float* b1      = (const float*)d_in[5];
    const float* gamma   = (const float*)d_in[6];
    const float* beta    = (const float*)d_in[7];
    const float* w2      = (const float*)d_in[8];
    const float* b2      = (const float*)d_in[9];

    float* ws     = (float*)d_ws;
    float* interp = ws + I_INTERP;
    float* H      = ws + I_H;
    float* gsum   = ws + I_SUM;
    float* gsumsq = ws + I_SUMSQ;
    float* scale  = ws + I_SCALE;
    float* shift  = ws + I_SHIFT;
    float* out    = (float*)d_out;

    zero_sums_kernel<<<1, 256, 0, stream>>>(gsum);  // zeros sum[128]+sumsq[128]
    knn_interp_kernel<<<B_ * (N1_ / 256), 256, 0, stream>>>(xyz1, xyz2, points2, interp);
    gemm1_bn_kernel<<<ROWS_ / (8 * 16), 256, 0, stream>>>(points1, interp, w1, b1, H, gsum, gsumsq);
    bn_stats_kernel<<<1, 128, 0, stream>>>(gsum, gsumsq, gamma, beta, scale, shift);
    gemm2_kernel<<<ROWS_ / (8 * 16), 256, 0, stream>>>(H, scale, shift, w2, b2, out);
}